// Phase2OptimizedInSARModel_85779086835982
// MI455X (gfx1250) — compile-verified
//
#include <hip/hip_runtime.h>
#include <math.h>

#define N_ST   50000
#define N_T    1024
#define N_K    8
#define TWO_PI 6.283185307179586f

typedef float v2f __attribute__((ext_vector_type(2)));
typedef float v8f __attribute__((ext_vector_type(8)));

// ---------------------------------------------------------------------------
// Kernel 1: per-time basis table Bt[t][12] (transposed for float2 K-loads)
//   rows: 1, t, sin/cos of 3 seasonal freqs, sin/cos of 2 long-term freqs
// ---------------------------------------------------------------------------
__global__ void build_basis_kernel(const float* __restrict__ time_vec,
                                   const float* __restrict__ lt_periods,
                                   float* __restrict__ Bt) {
  int t = blockIdx.x * blockDim.x + threadIdx.x;
  if (t >= N_T) return;
  float tv = time_vec[t];
  float* row = Bt + t * 12;
  row[0] = 1.0f;
  row[1] = tv;
  const float per[3] = {0.25f, 0.5f, 1.0f};
#pragma unroll
  for (int c = 0; c < 3; ++c) {
    float w = (TWO_PI / per[c]) * tv;
    row[2 + 2 * c]     = sinf(w);
    row[2 + 2 * c + 1] = cosf(w);
  }
#pragma unroll
  for (int c = 0; c < 2; ++c) {
    float w = (TWO_PI / lt_periods[c]) * tv;
    row[8 + 2 * c]     = sinf(w);
    row[8 + 2 * c + 1] = cosf(w);
  }
}

// ---------------------------------------------------------------------------
// Kernel 2: per-station coefficient matrix A[n][12]
//   Does the neighbor-smoothing gather. atan2 is algebraically eliminated:
//   amp*sin(wt+phi) = (amp*cos(phi))*sin(wt) + (amp*sin(phi))*cos(wt)
//   cos(atan2(mi,mr)) = mr / hypot(mr,mi)
// ---------------------------------------------------------------------------
__global__ void build_coeff_kernel(const float* __restrict__ offset,
                                   const float* __restrict__ trend,
                                   const float* __restrict__ samp,    // N x 3
                                   const float* __restrict__ sphase,  // N x 3
                                   const float* __restrict__ ltamp,   // N x 2
                                   const float* __restrict__ ltphase, // N x 2
                                   const float* __restrict__ sw,      // N
                                   const int*   __restrict__ nidx,    // N x 8
                                   const float* __restrict__ nw,      // N x 8
                                   float* __restrict__ A) {
  int n = blockIdx.x * blockDim.x + threadIdx.x;
  if (n >= N_ST) return;

  float mix  = 1.0f / (1.0f + expf(-sw[n]));   // sigmoid
  float omix = 1.0f - mix;

  float* a = A + n * 12;
  a[0] = offset[n];
  a[1] = trend[n];

  int   idx[N_K];
  float w[N_K];
#pragma unroll
  for (int k = 0; k < N_K; ++k) {
    idx[k] = nidx[n * N_K + k];
    w[k]   = nw[n * N_K + k];
  }

#pragma unroll
  for (int c = 0; c < 3; ++c) {
    float amp = samp[n * 3 + c];
    float ph  = sphase[n * 3 + c];
    float wavg = 0.0f, wr = 0.0f, wi = 0.0f;
#pragma unroll
    for (int k = 0; k < N_K; ++k) {
      int j = idx[k];
      wavg += w[k] * samp[j * 3 + c];
      float pj = sphase[j * 3 + c];
      wr += w[k] * cosf(pj);
      wi += w[k] * sinf(pj);
    }
    float sm_amp = omix * amp + mix * wavg;
    float mr = omix * cosf(ph) + mix * wr;
    float mi = omix * sinf(ph) + mix * wi;
    float r2 = mr * mr + mi * mi;
    float inv = (r2 > 0.0f) ? rsqrtf(r2) : 0.0f;
    float cphi = (r2 > 0.0f) ? mr * inv : 1.0f;  // atan2(0,0)=0 -> cos=1
    float sphi = (r2 > 0.0f) ? mi * inv : 0.0f;
    a[2 + 2 * c]     = sm_amp * cphi;  // multiplies sin(w t)
    a[2 + 2 * c + 1] = sm_amp * sphi;  // multiplies cos(w t)
  }

#pragma unroll
  for (int c = 0; c < 2; ++c) {
    float amp = ltamp[n * 2 + c];
    float ph  = ltphase[n * 2 + c];
    a[8 + 2 * c]     = amp * cosf(ph);
    a[8 + 2 * c + 1] = amp * sinf(ph);
  }
}

// ---------------------------------------------------------------------------
// Kernel 3: signals = A[N x 12] @ B[12 x T] via V_WMMA_F32_16X16X4_F32.
//   One wave per 16x16 output tile, 4 t-tiles per wave (A-frag reuse).
//   K=12 -> 3 WMMA issues per tile, exact fp32 accumulation.
//   A-frag lane layout (16x4 f32 A): lane l -> M = l%16,
//     VGPR0 = K(2*(l/16)), VGPR1 = K(2*(l/16)+1); B symmetric with N = l%16.
//   D layout: VGPR r, lane l -> row n0 + r + 8*(l/16), col t0 + l%16.
// ---------------------------------------------------------------------------
__global__ void wmma_signal_kernel(const float* __restrict__ A,
                                   const float* __restrict__ Bt,
                                   float* __restrict__ out) {
  const int wave   = (blockIdx.x * blockDim.x + threadIdx.x) >> 5;
  const int lane   = threadIdx.x & 31;
  const int n_tile = wave >> 4;   // 16 t-quads (of 4 tiles) per n-tile
  const int t_quad = wave & 15;
  if (n_tile >= (N_ST / 16)) return;   // wave-uniform; EXEC all-ones inside

  const int m  = lane & 15;
  const int kh = lane >> 4;
  const int n0 = n_tile * 16;

  // A fragments for the three K-steps (k = 0..3, 4..7, 8..11)
  const float* arow = A + (size_t)(n0 + m) * 12 + 2 * kh;
  v2f a0 = *(const v2f*)(arow + 0);
  v2f a1 = *(const v2f*)(arow + 4);
  v2f a2 = *(const v2f*)(arow + 8);

#pragma unroll
  for (int tt = 0; tt < 4; ++tt) {
    const int t_base = t_quad * 64 + tt * 16;
    const float* brow = Bt + (size_t)(t_base + m) * 12 + 2 * kh;
    v2f b0 = *(const v2f*)(brow + 0);
    v2f b1 = *(const v2f*)(brow + 4);
    v2f b2 = *(const v2f*)(brow + 8);

    v8f acc = {};
    acc = __builtin_amdgcn_wmma_f32_16x16x4_f32(false, a0, false, b0,
                                                (short)0, acc, false, false);
    acc = __builtin_amdgcn_wmma_f32_16x16x4_f32(false, a1, false, b1,
                                                (short)0, acc, false, false);
    acc = __builtin_amdgcn_wmma_f32_16x16x4_f32(false, a2, false, b2,
                                                (short)0, acc, false, false);

    float* orow = out + (size_t)(n0 + 8 * kh) * N_T + (t_base + m);
#pragma unroll
    for (int r = 0; r < 8; ++r)
      __builtin_nontemporal_store(acc[r], orow + (size_t)r * N_T);
  }
}

// ---------------------------------------------------------------------------
extern "C" void kernel_launch(void* const* d_in, const int* in_sizes, int n_in,
                              void* d_out, int out_size, void* d_ws, size_t ws_size,
                              hipStream_t stream) {
  const float* time_vec  = (const float*)d_in[0];
  const float* offset    = (const float*)d_in[1];
  const float* trend     = (const float*)d_in[2];
  const float* samp      = (const float*)d_in[3];
  const float* sphase    = (const float*)d_in[4];
  const float* ltamp     = (const float*)d_in[5];
  const float* ltphase   = (const float*)d_in[6];
  const float* ltper     = (const float*)d_in[7];
  const float* sw        = (const float*)d_in[8];
  const int*   nidx      = (const int*)d_in[9];
  const float* nw        = (const float*)d_in[10];
  float*       out       = (float*)d_out;

  // workspace: A = N_ST*12 floats (2.4 MB), then Bt = N_T*12 floats (48 KB)
  float* A  = (float*)d_ws;
  float* Bt = A + (size_t)N_ST * 12;

  build_basis_kernel<<<(N_T + 255) / 256, 256, 0, stream>>>(time_vec, ltper, Bt);
  build_coeff_kernel<<<(N_ST + 255) / 256, 256, 0, stream>>>(
      offset, trend, samp, sphase, ltamp, ltphase, sw, nidx, nw, A);

  // (N_ST/16) n-tiles * 16 t-quads = 50000 waves; 8 waves per 256-thread block
  const int total_waves = (N_ST / 16) * 16;
  const int blocks = (total_waves * 32 + 255) / 256;
  wmma_signal_kernel<<<blocks, 256, 0, stream>>>(A, Bt, out);
}